// ICONLoss_78237124264487
// MI455X (gfx1250) — compile-verified
//
#include <hip/hip_runtime.h>
#include <stdint.h>

#define DD    128
#define DMAX  127
#define NVOX  (DD*DD*DD)   // 2,097,152 = 8192 * 256 exactly
#define EPSF  1e-5f

typedef __attribute__((ext_vector_type(2))) float v2f;
typedef __attribute__((ext_vector_type(4))) float v4f;
typedef __attribute__((ext_vector_type(8))) float v8f;
typedef unsigned int u32;
typedef __attribute__((ext_vector_type(4))) u32 u32x4;
typedef __attribute__((ext_vector_type(8))) int i32x8;
typedef __attribute__((ext_vector_type(4))) int i32x4;

// ---------------- Gaussian taps (sigma=1, ksize=5, normalized) ----------------
__device__ __forceinline__ float kw(int o) {
  o = o < 0 ? -o : o;
  return o == 0 ? 0.40261996f : (o == 1 ? 0.24420135f : (o == 2 ? 0.05448868f : 0.0f));
}

// ---------------- block-wide sum -> atomicAdd (blockDim.x == 256) -------------
__device__ __forceinline__ void blockReduceAdd(float val, float* target) {
  __shared__ float sbuf[256];
  const int tid = threadIdx.x;
  sbuf[tid] = val;
  __syncthreads();
  for (int s = 128; s > 0; s >>= 1) {
    if (tid < s) sbuf[tid] += sbuf[tid + s];
    __syncthreads();
  }
  if (tid == 0) atomicAdd(target, sbuf[0]);
  __syncthreads();
}

// ---------------- trilinear helper -------------------------------------------
struct TriAxis { int i0, i1; float w0, w1, v0, v1; };
__device__ __forceinline__ TriAxis triAxis(float c) {
  TriAxis t;
  float f0 = floorf(c);
  int a = (int)f0;
  int b = a + 1;
  float f = c - f0;
  t.v0 = (a >= 0 && a <= DMAX) ? 1.f : 0.f;
  t.v1 = (b >= 0 && b <= DMAX) ? 1.f : 0.f;
  t.i0 = a < 0 ? 0 : (a > DMAX ? DMAX : a);
  t.i1 = b < 0 ? 0 : (b > DMAX ? DMAX : b);
  t.w0 = 1.f - f;
  t.w1 = f;
  return t;
}

// ---------------- 3-channel warp (optionally Ieps = grid + noise/128) ---------
template <bool IEPS>
__device__ __forceinline__ void warp3_gather(const float* __restrict__ phi,
                                             const float* __restrict__ src, int p,
                                             float& o0, float& o1, float& o2) {
  const float z = (phi[p] + 1.f) * 0.5f * 127.f;
  const float y = (phi[NVOX + p] + 1.f) * 0.5f * 127.f;
  const float x = (phi[2*NVOX + p] + 1.f) * 0.5f * 127.f;
  TriAxis tz = triAxis(z), ty = triAxis(y), tx = triAxis(x);
  int   zi[2] = {tz.i0, tz.i1}, yi[2] = {ty.i0, ty.i1}, xi[2] = {tx.i0, tx.i1};
  float zw[2] = {tz.w0*tz.v0, tz.w1*tz.v1};
  float yw[2] = {ty.w0*ty.v0, ty.w1*ty.v1};
  float xw[2] = {tx.w0*tx.v0, tx.w1*tx.v1};
  o0 = o1 = o2 = 0.f;
  #pragma unroll
  for (int dz = 0; dz < 2; ++dz)
  #pragma unroll
  for (int dy = 0; dy < 2; ++dy)
  #pragma unroll
  for (int dx = 0; dx < 2; ++dx) {
    const float w = zw[dz]*yw[dy]*xw[dx];
    const int zc = zi[dz], yc = yi[dy], xc = xi[dx];
    const int flat = (zc << 14) | (yc << 7) | xc;
    float s0 = src[flat], s1 = src[NVOX + flat], s2 = src[2*NVOX + flat];
    if (IEPS) {
      s0 = (-1.f + zc * (2.f/127.f)) + s0 * (1.f/128.f);
      s1 = (-1.f + yc * (2.f/127.f)) + s1 * (1.f/128.f);
      s2 = (-1.f + xc * (2.f/127.f)) + s2 * (1.f/128.f);
    }
    o0 += w*s0; o1 += w*s1; o2 += w*s2;
  }
}

__device__ __forceinline__ float warp1_gather(const float* __restrict__ phi,
                                              const float* __restrict__ src, int p) {
  const float z = (phi[p] + 1.f) * 0.5f * 127.f;
  const float y = (phi[NVOX + p] + 1.f) * 0.5f * 127.f;
  const float x = (phi[2*NVOX + p] + 1.f) * 0.5f * 127.f;
  TriAxis tz = triAxis(z), ty = triAxis(y), tx = triAxis(x);
  int   zi[2] = {tz.i0, tz.i1}, yi[2] = {ty.i0, ty.i1}, xi[2] = {tx.i0, tx.i1};
  float zw[2] = {tz.w0*tz.v0, tz.w1*tz.v1};
  float yw[2] = {ty.w0*ty.v0, ty.w1*ty.v1};
  float xw[2] = {tx.w0*tx.v0, tx.w1*tx.v1};
  float o = 0.f;
  #pragma unroll
  for (int dz = 0; dz < 2; ++dz)
  #pragma unroll
  for (int dy = 0; dy < 2; ++dy)
  #pragma unroll
  for (int dx = 0; dx < 2; ++dx) {
    const int flat = (zi[dz] << 14) | (yi[dy] << 7) | xi[dx];
    o += zw[dz]*yw[dy]*xw[dx] * src[flat];
  }
  return o;
}

// ---------------- kernels -----------------------------------------------------
template <bool IEPS>
__global__ __launch_bounds__(256) void warp3_kernel(const float* __restrict__ phi,
    const float* __restrict__ src, float* __restrict__ out) {
  const int p = blockIdx.x * 256 + threadIdx.x;
  float o0, o1, o2;
  warp3_gather<IEPS>(phi, src, p, o0, o1, o2);
  out[p] = o0; out[NVOX + p] = o1; out[2*NVOX + p] = o2;
}

// warp src3 by phi, compare against Ieps(p), accumulate squared error
__global__ __launch_bounds__(256) void warp3_mse_kernel(const float* __restrict__ phi,
    const float* __restrict__ src, const float* __restrict__ noise, float* acc) {
  const int p = blockIdx.x * 256 + threadIdx.x;
  float o0, o1, o2;
  warp3_gather<false>(phi, src, p, o0, o1, o2);
  const int pz = p >> 14, py = (p >> 7) & 127, px = p & 127;
  const float e0 = (-1.f + pz * (2.f/127.f)) + noise[p]          * (1.f/128.f);
  const float e1 = (-1.f + py * (2.f/127.f)) + noise[NVOX + p]   * (1.f/128.f);
  const float e2 = (-1.f + px * (2.f/127.f)) + noise[2*NVOX + p] * (1.f/128.f);
  const float d = (e0-o0)*(e0-o0) + (e1-o1)*(e1-o1) + (e2-o2)*(e2-o2);
  blockReduceAdd(d, acc);
}

__global__ __launch_bounds__(256) void warp1_kernel(const float* __restrict__ phi,
    const float* __restrict__ src, float* __restrict__ out) {
  const int p = blockIdx.x * 256 + threadIdx.x;
  out[p] = warp1_gather(phi, src, p);
}

// warp label, fuse dice partial sums: acc3[0]=sum(wl*other), [1]=sum(wl), [2]=sum(other)
__global__ __launch_bounds__(256) void warp1_dice_kernel(const float* __restrict__ phi,
    const float* __restrict__ src, const float* __restrict__ other, float* acc3) {
  const int p = blockIdx.x * 256 + threadIdx.x;
  const float wl = warp1_gather(phi, src, p);
  const float ot = other[p];
  blockReduceAdd(wl * ot, acc3 + 0);
  blockReduceAdd(wl,      acc3 + 1);
  blockReduceAdd(ot,      acc3 + 2);
}

__global__ __launch_bounds__(256) void products_kernel(const float* __restrict__ I,
    const float* __restrict__ J, float* __restrict__ pIJ, float* __restrict__ pII,
    float* __restrict__ pJJ) {
  const int p = blockIdx.x * 256 + threadIdx.x;
  const float a = I[p], b = J[p];
  pIJ[p] = a * b; pII[p] = a * a; pJJ[p] = b * b;
}

// ---- Tensor Data Mover: one-instruction 2-D tile DMA global -> LDS ----------
// D# per CDNA5 ISA. 2-D tile: groups 2/3 (and extra) zero.
__device__ __forceinline__ void tdm_load_2d(u32 lds_byte_addr, const float* gptr,
                                            u32 tile0, u32 tile1, u32 dim0_stride,
                                            u32 td0, u32 td1,
                                            u32 pad_en, u32 pad_intv, u32 pad_amt) {
  const uint64_t ga = (uint64_t)(uintptr_t)gptr;
  u32x4 g0;
  g0.x = 1u;                                             // count=1, user descriptor
  g0.y = lds_byte_addr;                                  // lds_addr [63:32]
  g0.z = (u32)(ga & 0xFFFFFFFFu);                        // global_addr[31:0]
  g0.w = (u32)((ga >> 32) & 0x01FFFFFFu) | (2u << 30);   // addr[56:32] | type=2
  i32x8 g1;
  g1[0] = (int)((2u << 16) | (pad_en << 20) | (pad_intv << 22) | (pad_amt << 25)); // data_size=4B
  g1[1] = (int)((td0 & 0xFFFFu) << 16);                  // tensor_dim0[15:0] @48
  g1[2] = (int)((td0 >> 16) | ((td1 & 0xFFFFu) << 16));  // td0[31:16] | td1[15:0]
  g1[3] = (int)((td1 >> 16) | (tile0 << 16));            // td1[31:16] | tile_dim0
  g1[4] = (int)(tile1 & 0xFFFFu);                        // tile_dim1; tile_dim2=0
  g1[5] = (int)dim0_stride;                              // tensor_dim0_stride[31:0]
  g1[6] = 0;                                             // stride[47:32], dim1_stride=0
  g1[7] = 0;
  const i32x4 z4 = {0, 0, 0, 0};
  const i32x8 z8 = {0, 0, 0, 0, 0, 0, 0, 0};
  __builtin_amdgcn_tensor_load_to_lds(g0, g1, z4, z4, z8, 0);
}

// ---- Gaussian blur along one axis as banded GEMM via V_WMMA_F32_16X16X4_F32 --
// Rows r = r1*128 + r0; element offset = r1*S1 + r0*S0 + w*elemStride.
// WPASS (elemStride==1): LDS rows of 160 (16 pad | 128 data | 16 pad), TDM pads.
// !WPASS (H/D):          LDS transposed planes lds[(k+4)*16 + row], k in [-4,132).
// Template parameter keeps the unrolled WMMA loop branch-free.
struct BlurArgs {
  const float* src[5];
  float*       dst[5];
  int S1, S0, elemStride;
};

template <bool WPASS>
__global__ __launch_bounds__(32) void blur_axis_wmma(BlurArgs args) {
  const int lane = threadIdx.x;             // one wave32 per block
  const float* __restrict__ src = args.src[blockIdx.y];
  float* __restrict__       dst = args.dst[blockIdx.y];
  const int t    = blockIdx.x;              // 0..1023 tiles of 16 rows
  const int r1   = t >> 3;
  const int r0b  = (t & 7) << 4;
  const int base = r1 * args.S1 + r0b * args.S0;

  __shared__ float lds[2560];               // max(16*160, 136*16)
  const u32 lds0 = (u32)(uintptr_t)&lds[0];

  if (WPASS) {
    // zero pad columns [0,16) and [144,160) of each 160-float row
    for (int i = lane; i < 512; i += 32) {
      const int row = i >> 5, q = i & 31;
      lds[row * 160 + (q < 16 ? q : 128 + q)] = 0.f;
    }
    // 16 rows x 128 contiguous floats; TDM inserts 32-dword pad every 128 dwords
    tdm_load_2d(lds0 + 16 * 4, src + base,
                /*tile0*/128, /*tile1*/16, /*dim0_stride*/(u32)args.S0,
                /*td0*/128, /*td1*/16, /*pad*/1, /*intv 128dw*/6, /*amt 32dw*/31);
  } else {
    // zero planes k+4 in [0,4) and [132,136)
    for (int i = lane; i < 128; i += 32) {
      const int pl = i >> 4, mm = i & 15;
      lds[(pl < 4 ? pl : 128 + pl) * 16 + mm] = 0.f;
    }
    // x = 16 contiguous row-elements, y = 128 positions along blur axis
    tdm_load_2d(lds0 + 4 * 16 * 4, src + base,
                /*tile0*/16, /*tile1*/128, /*dim0_stride*/(u32)args.elemStride,
                /*td0*/16, /*td1*/128, 0, 0, 0);
  }

  // Band-matrix B tiles (4x16), one per diagonal offset d = 4-4j, j=0..5.
  // B layout: VGPR v holds row K=v (lanes 0-15) / K=v+2 (lanes 16-31).
  const int nn = lane & 15;
  const int k0 = (lane < 16) ? 0 : 2;
  v2f bt[6];
  #pragma unroll
  for (int j = 0; j < 6; ++j) {
    const int d = 4 - 4 * j;
    bt[j].x = kw(d + nn - k0);
    bt[j].y = kw(d + nn - k0 - 1);
  }

  __builtin_amdgcn_s_wait_tensorcnt(0);
  __syncthreads();

  // A layout (16x4 f32): lanes 0-15 hold M=lane, K=kb+0/1; lanes 16-31 M=lane-16, K=kb+2/3.
  const int m      = lane & 15;
  const int rowAdd = (lane < 16) ? 0 : 8;

  for (int n = 0; n < 8; ++n) {
    v8f acc = {};
    #pragma unroll
    for (int j = 0; j < 6; ++j) {
      const int kb = 16 * n - 4 + 4 * j;    // K-tile base (band only)
      v2f a;
      if (WPASS) {
        a = *(const v2f*)&lds[m * 160 + 16 + k0 + kb];
      } else {
        a.x = lds[(kb + k0 + 4) * 16 + m];
        a.y = lds[(kb + k0 + 5) * 16 + m];
      }
      acc = __builtin_amdgcn_wmma_f32_16x16x4_f32(
          false, a, false, bt[j], (short)0, acc, false, false);
    }
    // D layout: lane<16 -> M=v, lane>=16 -> M=v+8; N = lane&15
    const int obase = base + (16 * n + nn) * args.elemStride;
    if (!WPASS) {                           // S0==1: 8 consecutive floats per lane
      v4f lo = {acc[0], acc[1], acc[2], acc[3]};
      v4f hi = {acc[4], acc[5], acc[6], acc[7]};
      *(v4f*)&dst[obase + rowAdd]     = lo;
      *(v4f*)&dst[obase + rowAdd + 4] = hi;
    } else {
      #pragma unroll
      for (int v = 0; v < 8; ++v) dst[obase + (v + rowAdd) * args.S0] = acc[v];
    }
  }
}

__global__ __launch_bounds__(256) void lncc_reduce_kernel(const float* __restrict__ bI,
    const float* __restrict__ bJ, const float* __restrict__ bIJ,
    const float* __restrict__ bII, const float* __restrict__ bJJ, float* acc) {
  const int p = blockIdx.x * 256 + threadIdx.x;
  const float mi = bI[p], mj = bJ[p];
  const float cross = bIJ[p] - mi * mj;
  float vi = bII[p] - mi * mi; vi = vi > 0.f ? vi : 0.f; vi += EPSF;
  float vj = bJJ[p] - mj * mj; vj = vj > 0.f ? vj : 0.f; vj += EPSF;
  blockReduceAdd(1.f - cross / sqrtf(vi * vj), acc);
}

__global__ void finalize_kernel(const float* __restrict__ acc, float* __restrict__ out) {
  const float Nf = (float)NVOX;
  const float ic  = (acc[0] + acc[1]) / (3.f * Nf);
  const float sim = (acc[8] + acc[9]) / Nf;
  const float dA  = 1.f - (2.f * acc[2] + EPSF) / (acc[3] + acc[4] + EPSF);
  const float dB  = 1.f - (2.f * acc[5] + EPSF) / (acc[7] + acc[6] + EPSF);
  out[0] = 128.f * ic + sim + dA + dB;
}

// ---------------- host orchestration ------------------------------------------
static void launch_blur3(const float* f0, const float* f1, const float* prod,
                         float* ping, float* pong, hipStream_t stream) {
  BlurArgs a;
  // pass 1: blur along W (rows (d,h))
  a.src[0] = f0; a.src[1] = f1;
  a.src[2] = prod; a.src[3] = prod + NVOX; a.src[4] = prod + 2*(size_t)NVOX;
  for (int i = 0; i < 5; ++i) a.dst[i] = ping + (size_t)i * NVOX;
  a.S1 = 16384; a.S0 = 128; a.elemStride = 1;
  blur_axis_wmma<true><<<dim3(1024, 5), 32, 0, stream>>>(a);
  // pass 2: blur along H (rows (d,w))
  for (int i = 0; i < 5; ++i) { a.src[i] = ping + (size_t)i*NVOX; a.dst[i] = pong + (size_t)i*NVOX; }
  a.S1 = 16384; a.S0 = 1; a.elemStride = 128;
  blur_axis_wmma<false><<<dim3(1024, 5), 32, 0, stream>>>(a);
  // pass 3: blur along D (rows (h,w))
  for (int i = 0; i < 5; ++i) { a.src[i] = pong + (size_t)i*NVOX; a.dst[i] = ping + (size_t)i*NVOX; }
  a.S1 = 128; a.S0 = 1; a.elemStride = 16384;
  blur_axis_wmma<false><<<dim3(1024, 5), 32, 0, stream>>>(a);
}

extern "C" void kernel_launch(void* const* d_in, const int* in_sizes, int n_in,
                              void* d_out, int out_size, void* d_ws, size_t ws_size,
                              hipStream_t stream) {
  (void)in_sizes; (void)n_in; (void)out_size; (void)ws_size;
  const float* phiAB  = (const float*)d_in[0];
  const float* phiBA  = (const float*)d_in[1];
  const float* imageA = (const float*)d_in[2];
  const float* imageB = (const float*)d_in[3];
  const float* labelA = (const float*)d_in[4];
  const float* labelB = (const float*)d_in[5];
  const float* noise  = (const float*)d_in[6];
  float* out = (float*)d_out;
  float* ws  = (float*)d_ws;

  float* acc  = ws;                              // 64 floats of accumulators
  float* Ibuf = ws + 64;                         // N
  float* prod = Ibuf + (size_t)NVOX;             // 3N (IJ, II, JJ)
  float* ping = prod + 3*(size_t)NVOX;           // 5N
  float* pong = ping + 5*(size_t)NVOX;           // 5N
  float* tmp3 = pong;                            // aliases pong; used before LNCC

  (void)hipMemsetAsync(acc, 0, 64 * sizeof(float), stream);

  const dim3 g(NVOX / 256), b(256);

  // inverse-consistency losses (Ieps computed analytically at gather time)
  warp3_kernel<true><<<g, b, 0, stream>>>(phiAB, noise, tmp3);
  warp3_mse_kernel<<<g, b, 0, stream>>>(phiBA, tmp3, noise, acc + 0);
  warp3_kernel<true><<<g, b, 0, stream>>>(phiBA, noise, tmp3);
  warp3_mse_kernel<<<g, b, 0, stream>>>(phiAB, tmp3, noise, acc + 1);

  // dice terms (fully fused, no storage)
  warp1_dice_kernel<<<g, b, 0, stream>>>(phiAB, labelA, labelB, acc + 2);
  warp1_dice_kernel<<<g, b, 0, stream>>>(phiBA, labelB, labelA, acc + 5);

  // lncc(warp(image_A, phiAB), image_B)
  warp1_kernel<<<g, b, 0, stream>>>(phiAB, imageA, Ibuf);
  products_kernel<<<g, b, 0, stream>>>(Ibuf, imageB, prod, prod + NVOX, prod + 2*(size_t)NVOX);
  launch_blur3(Ibuf, imageB, prod, ping, pong, stream);
  lncc_reduce_kernel<<<g, b, 0, stream>>>(ping, ping + NVOX, ping + 2*(size_t)NVOX,
                                          ping + 3*(size_t)NVOX, ping + 4*(size_t)NVOX, acc + 8);

  // lncc(image_A, warp(image_B, phiBA))
  warp1_kernel<<<g, b, 0, stream>>>(phiBA, imageB, Ibuf);
  products_kernel<<<g, b, 0, stream>>>(imageA, Ibuf, prod, prod + NVOX, prod + 2*(size_t)NVOX);
  launch_blur3(imageA, Ibuf, prod, ping, pong, stream);
  lncc_reduce_kernel<<<g, b, 0, stream>>>(ping, ping + NVOX, ping + 2*(size_t)NVOX,
                                          ping + 3*(size_t)NVOX, ping + 4*(size_t)NVOX, acc + 9);

  finalize_kernel<<<1, 1, 0, stream>>>(acc, out);
}